// LocalMatching_42107859370946
// MI455X (gfx1250) — compile-verified
//
#include <hip/hip_runtime.h>
#include <math.h>

typedef float v2f __attribute__((ext_vector_type(2)));
typedef float v8f __attribute__((ext_vector_type(8)));

#define PAD_IDX 1
#define EPS_F 1e-8f
#define NT 4   // N-tiles per wave (A fragment reuse)

// ---------------------------------------------------------------------------
// Kernel 1: embedding gather + PAD mask + row L2 norm
//   out_e[r][0:64] = table[ids[r]] * (ids[r] != PAD)
//   out_n[r]       = ||out_e[r]||
// ---------------------------------------------------------------------------
__global__ void lm_gather_mask_norm(const float* __restrict__ table,
                                    const int* __restrict__ ids,
                                    float* __restrict__ out_e,
                                    float* __restrict__ out_n,
                                    int n_rows) {
    int r = blockIdx.x * blockDim.x + threadIdx.x;
    if (r >= n_rows) return;
    int id = ids[r];
    float mask = (id != PAD_IDX) ? 1.0f : 0.0f;
    const float4* __restrict__ src = (const float4*)(table + (size_t)id * 64);
    float4* __restrict__ dst = (float4*)(out_e + (size_t)r * 64);
    float s = 0.0f;
#pragma unroll
    for (int i = 0; i < 16; ++i) {
        float4 v = src[i];
        v.x *= mask; v.y *= mask; v.z *= mask; v.w *= mask;
        s += v.x * v.x + v.y * v.y + v.z * v.z + v.w * v.w;
        dst[i] = v;
    }
    out_n[r] = sqrtf(s);
}

// ---------------------------------------------------------------------------
// Kernel 2: WMMA fp32 GEMM (qe . de^T), normalize, xor, permuted scatter.
// One wave handles one 16x16 M/N tile, for NT consecutive N-tiles.
//
// fp32 16x4 A fragment layout (ISA 7.12.2):
//   lane = m + 16*(k>=2), vgpr = k&1  -> lane holds A[m][kbase+koff .. +1],
//   a contiguous float2 of the qe row.  B (4x16, = de^T) is symmetric in n.
// C/D: vgpr r holds M = r + 8*(lane>=16), N = lane&15.
// ---------------------------------------------------------------------------
__global__ void __launch_bounds__(256)
lm_match_wmma(const float* __restrict__ de, const float* __restrict__ dn,
              const float* __restrict__ qe, const float* __restrict__ qn,
              const int* __restrict__ q_ids, const int* __restrict__ d_ids,
              const int* __restrict__ para_ptr,
              float* __restrict__ out,
              int Lq, int D, int Mtiles, int Ntiles, int totalWaves) {
    const int lane = threadIdx.x & 31;
    const int wid  = (int)((blockIdx.x * blockDim.x + threadIdx.x) >> 5);
    if (wid >= totalWaves) return;                 // wave-uniform

    // mtile varies fastest -> concurrent waves share the same B columns (L2 reuse)
    const int mtile = wid % Mtiles;
    const int nbase = (wid / Mtiles) * NT;

    const int para     = *para_ptr;                // uniform scalar load
    const int num_para = D / para;
    const unsigned nqd = (unsigned)num_para * (unsigned)para * (unsigned)Lq;

    const int half = lane >> 4;                    // 0: lanes 0-15, 1: lanes 16-31
    const int mrow = lane & 15;
    const int koff = half * 2;

    // A fragments for the whole K=64 strip (kept in registers across N-tiles)
    v2f afrag[16];
    const float* __restrict__ aptr =
        qe + ((size_t)(mtile * 16 + mrow) * 64 + koff);
#pragma unroll
    for (int kk = 0; kk < 16; ++kk)
        afrag[kk] = *(const v2f*)(aptr + 4 * kk);

    // Per-lane M-row metadata: accumulator vgpr r -> row q = qbase + r
    const int qbase = mtile * 16 + half * 8;
    float4 qnA = *(const float4*)(qn + qbase);
    float4 qnB = *(const float4*)(qn + qbase + 4);
    int4   qiA = *(const int4*)(q_ids + qbase);
    int4   qiB = *(const int4*)(q_ids + qbase + 4);
    const float qnv[8] = {qnA.x, qnA.y, qnA.z, qnA.w, qnB.x, qnB.y, qnB.z, qnB.w};
    const int   qiv[8] = {qiA.x, qiA.y, qiA.z, qiA.w, qiB.x, qiB.y, qiB.z, qiB.w};

    for (int t = 0; t < NT; ++t) {
        const int ntile = nbase + t;
        if (ntile >= Ntiles) break;                // wave-uniform

        const int dcol = ntile * 16 + (lane & 15);
        const float* __restrict__ bptr = de + ((size_t)dcol * 64 + koff);

        v2f bfrag[16];
#pragma unroll
        for (int kk = 0; kk < 16; ++kk)
            bfrag[kk] = *(const v2f*)(bptr + 4 * kk);

        v8f acc = {};
#pragma unroll
        for (int kk = 0; kk < 16; ++kk)
            acc = __builtin_amdgcn_wmma_f32_16x16x4_f32(
                false, afrag[kk], false, bfrag[kk],
                (short)0, acc, false, false);

        const float dnv = dn[dcol];
        const int   did = d_ids[dcol];

        // Output permutation: d -> (p = d/para, j = d%para); per-lane constant.
        const unsigned p  = (unsigned)dcol / (unsigned)para;
        const unsigned j  = (unsigned)dcol - p * (unsigned)para;
        const unsigned c0 = p * (unsigned)(Lq * para) + j;

#pragma unroll
        for (int r = 0; r < 8; ++r) {
            const int   q     = qbase + r;
            const float denom = fmaxf(qnv[r] * dnv, EPS_F);
            const float cosv  = acc[r] / denom;
            const float xorv  = (qiv[r] == did) ? 1.0f : 0.0f;
            const unsigned idx = c0 + (unsigned)q * (unsigned)para;
            out[idx]       = xorv;   // xor_matrix (flattened, float)
            out[nqd + idx] = cosv;   // cos_matrix
        }
    }
}

// ---------------------------------------------------------------------------
extern "C" void kernel_launch(void* const* d_in, const int* in_sizes, int n_in,
                              void* d_out, int out_size, void* d_ws, size_t ws_size,
                              hipStream_t stream) {
    const float* table    = (const float*)d_in[0];
    const int*   q_ids    = (const int*)d_in[1];
    const int*   d_ids    = (const int*)d_in[2];
    const int*   para_ptr = (const int*)d_in[3];

    const int Lq = in_sizes[1];   // 512   (512x1 ids)
    const int D  = in_sizes[2];   // 100000

    // Workspace layout (floats): de[D*64] | dn[D] | qe[Lq*64] | qn[Lq]  (~26 MB)
    float* ws = (float*)d_ws;
    float* de = ws;
    float* dn = de + (size_t)D * 64;
    float* qe = dn + D;
    float* qn = qe + (size_t)Lq * 64;

    lm_gather_mask_norm<<<(D + 255) / 256, 256, 0, stream>>>(table, d_ids, de, dn, D);
    lm_gather_mask_norm<<<(Lq + 255) / 256, 256, 0, stream>>>(table, q_ids, qe, qn, Lq);

    const int Mtiles = Lq / 16;                    // 32
    const int Ntiles = D / 16;                     // 6250
    const int jobsN  = (Ntiles + NT - 1) / NT;     // 1563
    const int totalWaves = Mtiles * jobsN;         // 50016
    const int blocks = (totalWaves + 7) / 8;       // 8 waves (256 thr) per block

    lm_match_wmma<<<blocks, 256, 0, stream>>>(de, dn, qe, qn, q_ids, d_ids, para_ptr,
                                              (float*)d_out, Lq, D, Mtiles, Ntiles,
                                              totalWaves);
}